// LearnableFilter_1778116461264
// MI455X (gfx1250) — compile-verified
//
#include <hip/hip_runtime.h>

// LearnableFilter: m_{t+1} = b_t*m_t + (1-b_t)*x_t, y_t = m_t (pre-update).
// Blocked-scan formulation: per 16-step chunk, y_tile(16t x 16b) = L @ x_tile + P*carry,
// computed with V_WMMA_F32_16X16X4_F32 (K accumulated over 4 wmma ops).
// Streaming data (x in, y out) uses non-temporal hints so the 128-way-reused
// L/P coefficient blocks stay resident in WGP$/L2.

#define N_F    32
#define NB_    2048
#define T_     1024
#define DT_    0.1f
#define RMIN_  1.0e5f
#define RMAX_  1.0e7f
#define CMIN_  1.0e-7f
#define CMAX_  1.0e-4f
#define CHUNK  16
#define NCHUNK (T_ / CHUNK)        /* 64 */
#define ABLK   288                 /* floats per (chunk,n) coeff block: 256 L + 16 P + b15,1-b15 + pad */

typedef __attribute__((ext_vector_type(2))) float v2f;
typedef __attribute__((ext_vector_type(4))) float v4f;
typedef __attribute__((ext_vector_type(8))) float v8f;

// ---------------- coefficient precompute: one thread per (chunk, n) ----------------
__global__ void build_coeffs(const float* __restrict__ Rl, const float* __restrict__ Cl,
                             const float* __restrict__ noise_R, const float* __restrict__ noise_C,
                             const float* __restrict__ mu, float* __restrict__ ws) {
  int tid = blockIdx.x * blockDim.x + threadIdx.x;   // 0 .. N_F*NCHUNK-1 (2048)
  if (tid >= N_F * NCHUNK) return;
  int n  = tid & (N_F - 1);
  int cI = tid >> 5;

  float R_true = (1.f / (1.f + __expf(-Rl[0]))) * (RMAX_ - RMIN_) + RMIN_;
  float C_true = (1.f / (1.f + __expf(-Cl[0]))) * (CMAX_ - CMIN_) + CMIN_;

  float b[16];
  #pragma unroll
  for (int j = 0; j < 16; ++j) {
    int t   = cI * CHUNK + j;
    int idx = t * N_F + n;                            // (T,N,1) layout
    float R  = R_true * noise_R[idx];
    float C  = C_true * noise_C[idx];
    float rc = mu[idx] * R * C;
    b[j] = rc / (rc + DT_);
  }

  float* Lp = ws + (size_t)(cI * N_F + n) * ABLK;
  #pragma unroll
  for (int i = 0; i < 16; ++i) {
    float row[16];
    #pragma unroll
    for (int j = 0; j < 16; ++j) row[j] = 0.f;
    float g = 1.f;                                    // running product b_{j+1}..b_{i-1}
    for (int j = i - 1; j >= 0; --j) { row[j] = (1.f - b[j]) * g; g *= b[j]; }
    #pragma unroll
    for (int j = 0; j < 16; ++j) Lp[i * 16 + j] = row[j];
    Lp[256 + i] = g;                                  // P_i = prod_{j<i} b_j
  }
  Lp[272] = b[15];
  Lp[273] = 1.f - b[15];
}

// ---------------- main scan: one wave per (n, 16-wide batch strip) ----------------
__global__ __launch_bounds__(256)
void filter_scan_wmma(const float* __restrict__ x, const float* __restrict__ mem0,
                      const float* __restrict__ ws, float* __restrict__ out) {
  const int lane = threadIdx.x & 31;
  const int wave = blockIdx.x * (blockDim.x >> 5) + (threadIdx.x >> 5);  // 0..4095
  const int STRIPS = NB_ / 16;                        // 128
  const int n   = wave / STRIPS;
  const int s   = wave - n * STRIPS;
  const int col = lane & 15;                          // A: M row; B/D: batch column
  const int hi  = lane >> 4;

  const int row = n * NB_ + s * 16 + col;
  const float* __restrict__ xrow = x   + (size_t)row * T_;
  float*       __restrict__ orow = out + (size_t)row * T_;
  float c = mem0[row];                                // carry m_{t0}

  for (int cI = 0; cI < NCHUNK; ++cI) {
    const int t0 = cI * CHUNK;
    const float* Ab = ws + (size_t)(cI * N_F + n) * ABLK;

    // A fragments (temporal: reused by 128 waves per n): lane holds L[col][4g + 2*hi + {0,1}]
    v2f a0 = *(const v2f*)(Ab + col * 16 + 0  + 2 * hi);
    v2f a1 = *(const v2f*)(Ab + col * 16 + 4  + 2 * hi);
    v2f a2 = *(const v2f*)(Ab + col * 16 + 8  + 2 * hi);
    v2f a3 = *(const v2f*)(Ab + col * 16 + 12 + 2 * hi);
    // P fragment: this lane's 8 output rows are M = v + 8*hi
    v4f pA = *(const v4f*)(Ab + 256 + 8 * hi);
    v4f pB = *(const v4f*)(Ab + 256 + 8 * hi + 4);
    float b15  = Ab[272];
    float ib15 = Ab[273];

    // B fragments (non-temporal: x is streamed, read exactly once):
    // lane holds x[t0 + 4g + 2*hi + {0,1}] of its batch column
    v2f b0 = __builtin_nontemporal_load((const v2f*)(xrow + t0 + 0  + 2 * hi));
    v2f b1 = __builtin_nontemporal_load((const v2f*)(xrow + t0 + 4  + 2 * hi));
    v2f b2 = __builtin_nontemporal_load((const v2f*)(xrow + t0 + 8  + 2 * hi));
    v2f b3 = __builtin_nontemporal_load((const v2f*)(xrow + t0 + 12 + 2 * hi));

    v8f acc = {0.f, 0.f, 0.f, 0.f, 0.f, 0.f, 0.f, 0.f};
    acc = __builtin_amdgcn_wmma_f32_16x16x4_f32(false, a0, false, b0, (short)0, acc, false, false);
    acc = __builtin_amdgcn_wmma_f32_16x16x4_f32(false, a1, false, b1, (short)0, acc, false, false);
    acc = __builtin_amdgcn_wmma_f32_16x16x4_f32(false, a2, false, b2, (short)0, acc, false, false);
    acc = __builtin_amdgcn_wmma_f32_16x16x4_f32(false, a3, false, b3, (short)0, acc, false, false);

    // carry contribution: y[i] += P_i * c
    acc[0] += pA[0] * c; acc[1] += pA[1] * c; acc[2] += pA[2] * c; acc[3] += pA[3] * c;
    acc[4] += pB[0] * c; acc[5] += pB[1] * c; acc[6] += pB[2] * c; acc[7] += pB[3] * c;

    // store (non-temporal: y is write-once streamed):
    // this lane's acc = y[t0+8*hi .. t0+8*hi+7] of its column (contiguous in t)
    float* op = orow + t0 + 8 * hi;
    __builtin_nontemporal_store((v4f){acc[0], acc[1], acc[2], acc[3]}, (v4f*)(op));
    __builtin_nontemporal_store((v4f){acc[4], acc[5], acc[6], acc[7]}, (v4f*)(op + 4));

    // carry update: m_{t0+16} = b15 * y[15] + (1-b15) * x[t0+15]
    // (valid on hi lanes: acc[7] = M=15, b3[1] = t0+15); broadcast to both halves
    float cn = b15 * acc[7] + ib15 * b3[1];
    c = __shfl(cn, 16 + col, 32);
  }
}

// ---------------- fallback (workspace too small): straight VALU scan ----------------
__global__ void filter_scan_fallback(const float* __restrict__ x,
                                     const float* __restrict__ Rl, const float* __restrict__ Cl,
                                     const float* __restrict__ noise_R, const float* __restrict__ noise_C,
                                     const float* __restrict__ mu, const float* __restrict__ mem0,
                                     float* __restrict__ out) {
  int row = blockIdx.x * blockDim.x + threadIdx.x;   // (n,b) row
  if (row >= N_F * NB_) return;
  int n = row / NB_;
  float R_true = (1.f / (1.f + __expf(-Rl[0]))) * (RMAX_ - RMIN_) + RMIN_;
  float C_true = (1.f / (1.f + __expf(-Cl[0]))) * (CMAX_ - CMIN_) + CMIN_;
  const float* xr = x + (size_t)row * T_;
  float* orr = out + (size_t)row * T_;
  float m = mem0[row];
  for (int t = 0; t < T_; ++t) {
    int idx = t * N_F + n;
    float rc = mu[idx] * (R_true * noise_R[idx]) * (C_true * noise_C[idx]);
    float b  = rc / (rc + DT_);
    orr[t] = m;
    m = b * m + (1.f - b) * xr[t];
  }
}

extern "C" void kernel_launch(void* const* d_in, const int* in_sizes, int n_in,
                              void* d_out, int out_size, void* d_ws, size_t ws_size,
                              hipStream_t stream) {
  (void)in_sizes; (void)n_in; (void)out_size;
  const float* x  = (const float*)d_in[0];
  const float* Rl = (const float*)d_in[1];
  const float* Cl = (const float*)d_in[2];
  const float* nR = (const float*)d_in[3];
  const float* nC = (const float*)d_in[4];
  const float* mu = (const float*)d_in[5];
  const float* m0 = (const float*)d_in[6];
  float* out = (float*)d_out;

  const size_t ws_needed = (size_t)N_F * NCHUNK * ABLK * sizeof(float);  // ~2.4 MB
  if (ws_size >= ws_needed) {
    float* ws = (float*)d_ws;
    build_coeffs<<<(N_F * NCHUNK + 255) / 256, 256, 0, stream>>>(Rl, Cl, nR, nC, mu, ws);
    // 4096 waves total, 8 waves (256 threads) per block
    filter_scan_wmma<<<(N_F * (NB_ / 16)) / 8, 256, 0, stream>>>(x, m0, ws, out);
  } else {
    filter_scan_fallback<<<(N_F * NB_ + 255) / 256, 256, 0, stream>>>(
        x, Rl, Cl, nR, nC, mu, m0, out);
  }
}